// SMPL_80504866996306
// MI455X (gfx1250) — compile-verified
//
#include <hip/hip_runtime.h>
#include <hip/hip_bf16.h>

#define B_  1024
#define V_  6890
#define V3_ 20670
#define J_  24

typedef float v2f __attribute__((ext_vector_type(2)));
typedef float v8f __attribute__((ext_vector_type(8)));

// ---------------- workspace layout (float offsets) ----------------
#define RS_OFF    0                       // [B][24][9]   = 221184
#define PF_OFF    221184                  // [B][208]     = 212992 (col 207 zero-padded)
#define SP12_OFF  434176                  // [B][12]      = 12288  (cols 10,11 zero)
#define VP_OFF    446464                  // [B][20670]   = 21166080
#define SDJ_OFF   21612544                // [10][72]     = 720
#define TJ_OFF    21613264                // [72]
#define AK_OFF    21613336                // [24*16][B]   = 393216
// total ~22.0M floats (~88 MB)

// ---------------- 1) Rodrigues + pose_feature (+ padded copies) ----------------
__global__ void rodrigues_kernel(const float* __restrict__ pose_params,
                                 const float* __restrict__ shape_params,
                                 float* __restrict__ Rs,
                                 float* __restrict__ pf,      // stride 208
                                 float* __restrict__ sp12) {  // stride 12
  int i = blockIdx.x * 256 + threadIdx.x;
  if (i >= B_ * J_) return;
  int b = i / J_, j = i % J_;
  float x = pose_params[b * 72 + j * 3 + 0];
  float y = pose_params[b * 72 + j * 3 + 1];
  float z = pose_params[b * 72 + j * 3 + 2];
  float px = x + 1e-8f, py = y + 1e-8f, pz = z + 1e-8f;
  float ang = sqrtf(px * px + py * py + pz * pz);
  float inv = 1.0f / ang;
  float rx = x * inv, ry = y * inv, rz = z * inv;
  float c = cosf(ang), s = sinf(ang), o = 1.0f - c;
  float R[9];
  R[0] = c + o * rx * rx;      R[1] = o * rx * ry - s * rz; R[2] = o * rx * rz + s * ry;
  R[3] = o * rx * ry + s * rz; R[4] = c + o * ry * ry;      R[5] = o * ry * rz - s * rx;
  R[6] = o * rx * rz - s * ry; R[7] = o * ry * rz + s * rx; R[8] = c + o * rz * rz;
  float* out = Rs + i * 9;
#pragma unroll
  for (int q = 0; q < 9; ++q) out[q] = R[q];
  if (j > 0) {
    float* p = pf + b * 208 + (j - 1) * 9;
#pragma unroll
    for (int q = 0; q < 9; ++q)
      p[q] = R[q] - ((q == 0 || q == 4 || q == 8) ? 1.0f : 0.0f);
  } else {
    pf[b * 208 + 207] = 0.0f;  // K-pad for WMMA
#pragma unroll
    for (int k = 0; k < 10; ++k) sp12[b * 12 + k] = shape_params[b * 10 + k];
    sp12[b * 12 + 10] = 0.0f;
    sp12[b * 12 + 11] = 0.0f;
  }
}

// ---------------- 2) fold J_regressor through shapedirs/template ----------------
__global__ void sdj_kernel(const float* __restrict__ shapedirs,
                           const float* __restrict__ v_template,
                           const float* __restrict__ J_regressor,
                           float* __restrict__ SD_J, float* __restrict__ T_J) {
  __shared__ float lds[256 * 11];
  const int jc = blockIdx.x;            // 0..71  (j*3+c)
  const int j = jc / 3, c = jc % 3;
  const int t = threadIdx.x;
  float acc[11];
#pragma unroll
  for (int k = 0; k < 11; ++k) acc[k] = 0.0f;
  for (int v = t; v < V_; v += 256) {
    float jr = J_regressor[v * J_ + j];
#pragma unroll
    for (int k = 0; k < 10; ++k)
      acc[k] += shapedirs[k * V3_ + v * 3 + c] * jr;
    acc[10] += v_template[v * 3 + c] * jr;
  }
#pragma unroll
  for (int k = 0; k < 11; ++k) lds[t * 11 + k] = acc[k];
  __syncthreads();
  for (int s = 128; s > 0; s >>= 1) {
    if (t < s)
#pragma unroll
      for (int k = 0; k < 11; ++k) lds[t * 11 + k] += lds[(t + s) * 11 + k];
    __syncthreads();
  }
  if (t == 0) {
#pragma unroll
    for (int k = 0; k < 10; ++k) SD_J[k * 72 + jc] = lds[k];
    T_J[jc] = lds[10];
  }
}

// ---------------- 3) v_posed GEMM via V_WMMA_F32_16X16X4_F32 ----------------
// C[16b x 16n]/wave; block = 8 waves => 16 x 128 tile. Branch-free inner loops:
// A operands are zero-padded in K, B row indices clamped (zero-A kills products),
// column index clamped (store is bounds-checked).
__global__ void vposed_gemm_kernel(const float* __restrict__ sp12,
                                   const float* __restrict__ pf,
                                   const float* __restrict__ shapedirs,
                                   const float* __restrict__ posedirs,
                                   const float* __restrict__ v_template,
                                   float* __restrict__ v_posed) {
  const int lane = threadIdx.x & 31;
  const int wave = threadIdx.x >> 5;
  const int n0 = blockIdx.x * 128 + wave * 16;
  const int m0 = blockIdx.y * 16;
  const int half = lane >> 4;
  const int l16 = lane & 15;
  const int mA = m0 + l16;
  const int nB = n0 + l16;
  const bool nOK = nB < V3_;
  const int nBc = nOK ? nB : (V3_ - 1);  // clamped column for loads

  v8f acc = {0.f, 0.f, 0.f, 0.f, 0.f, 0.f, 0.f, 0.f};

  // shapedirs contribution (K = 10, A padded to 12, B rows clamped)
#pragma unroll
  for (int k0 = 0; k0 < 12; k0 += 4) {
    int ka = k0 + 2 * half;
    int r0 = (ka < 10) ? ka : 9;
    int r1 = (ka + 1 < 10) ? (ka + 1) : 9;
    v2f a, b;
    a.x = sp12[mA * 12 + ka];
    a.y = sp12[mA * 12 + ka + 1];
    b.x = shapedirs[r0 * V3_ + nBc];
    b.y = shapedirs[r1 * V3_ + nBc];
    acc = __builtin_amdgcn_wmma_f32_16x16x4_f32(false, a, false, b, (short)0,
                                                acc, false, false);
  }
  // posedirs contribution (K = 207, A padded to 208, B rows clamped)
#pragma unroll 4
  for (int k0 = 0; k0 < 208; k0 += 4) {
    int ka = k0 + 2 * half;
    int r0 = (ka < 207) ? ka : 206;
    int r1 = (ka + 1 < 207) ? (ka + 1) : 206;
    v2f a, b;
    a.x = pf[mA * 208 + ka];
    a.y = pf[mA * 208 + ka + 1];
    b.x = posedirs[r0 * V3_ + nBc];
    b.y = posedirs[r1 * V3_ + nBc];
    acc = __builtin_amdgcn_wmma_f32_16x16x4_f32(false, a, false, b, (short)0,
                                                acc, false, false);
  }
  if (nOK) {
    float tmpl = v_template[nB];
#pragma unroll
    for (int r = 0; r < 8; ++r) {
      int m = m0 + r + 8 * half;
      v_posed[m * V3_ + nB] = acc[r] + tmpl;
    }
  }
}

// ---------------- 4) kinematic chain, per-batch ----------------
__global__ void rigid_kernel(const float* __restrict__ shape_params,
                             const float* __restrict__ Rs,
                             const float* __restrict__ SD_J,
                             const float* __restrict__ T_J,
                             const int* __restrict__ parents,
                             float* __restrict__ A_kcb,
                             float* __restrict__ J_out) {
  __shared__ float res[24][16];
  __shared__ float jl[24][3];
  const int b = blockIdx.x;
  const int j = threadIdx.x;
  float H[16];
  if (j < J_) {
    float c0 = T_J[j * 3 + 0], c1 = T_J[j * 3 + 1], c2 = T_J[j * 3 + 2];
#pragma unroll
    for (int k = 0; k < 10; ++k) {
      float s = shape_params[b * 10 + k];
      c0 += s * SD_J[k * 72 + j * 3 + 0];
      c1 += s * SD_J[k * 72 + j * 3 + 1];
      c2 += s * SD_J[k * 72 + j * 3 + 2];
    }
    jl[j][0] = c0; jl[j][1] = c1; jl[j][2] = c2;
  }
  __syncthreads();
  if (j < J_) {
    const float* R = Rs + (b * J_ + j) * 9;
    int p = parents[j];
    float t0 = jl[j][0], t1 = jl[j][1], t2 = jl[j][2];
    if (j > 0) { t0 -= jl[p][0]; t1 -= jl[p][1]; t2 -= jl[p][2]; }
    H[0] = R[0]; H[1] = R[1]; H[2] = R[2]; H[3] = t0;
    H[4] = R[3]; H[5] = R[4]; H[6] = R[5]; H[7] = t1;
    H[8] = R[6]; H[9] = R[7]; H[10] = R[8]; H[11] = t2;
    H[12] = 0.f; H[13] = 0.f; H[14] = 0.f; H[15] = 1.f;
  }
  __syncthreads();
  for (int i = 0; i < J_; ++i) {
    if (j == i) {
      if (i == 0) {
#pragma unroll
        for (int q = 0; q < 16; ++q) res[0][q] = H[q];
      } else {
        int p = parents[i];
#pragma unroll
        for (int r = 0; r < 4; ++r)
#pragma unroll
          for (int c = 0; c < 4; ++c) {
            float s = 0.f;
#pragma unroll
            for (int q = 0; q < 4; ++q) s += res[p][r * 4 + q] * H[q * 4 + c];
            res[i][r * 4 + c] = s;
          }
      }
    }
    __syncthreads();
  }
  if (j < J_) {
    J_out[(b * J_ + j) * 3 + 0] = res[j][3];
    J_out[(b * J_ + j) * 3 + 1] = res[j][7];
    J_out[(b * J_ + j) * 3 + 2] = res[j][11];
    float jx = jl[j][0], jy = jl[j][1], jz = jl[j][2];
#pragma unroll
    for (int r = 0; r < 4; ++r) {
      float adj = res[j][r * 4 + 0] * jx + res[j][r * 4 + 1] * jy +
                  res[j][r * 4 + 2] * jz;
#pragma unroll
      for (int c = 0; c < 4; ++c) {
        float val = res[j][r * 4 + c] - ((c == 3) ? adj : 0.f);
        A_kcb[(j * 16 + r * 4 + c) * B_ + b] = val;  // transposed for WMMA A
      }
    }
  }
}

// ---------------- 5) LBS skinning via WMMA ----------------
// Per wave: 16 batches x 16 vertices; 12 accumulators (T rows 0..2) x 6 K-steps.
__global__ void skin_kernel(const float* __restrict__ A_kcb,
                            const float* __restrict__ weights,
                            const float* __restrict__ v_posed,
                            float* __restrict__ verts) {
  const int lane = threadIdx.x & 31;
  const int wave = threadIdx.x >> 5;
  const int n0 = blockIdx.x * 128 + wave * 16;  // vertex tile
  const int m0 = blockIdx.y * 16;               // batch tile
  const int half = lane >> 4;
  const int l16 = lane & 15;
  const int v = n0 + l16;
  const bool vOK = v < V_;
  const int vc = vOK ? v : (V_ - 1);            // clamped for loads
  const int bA = m0 + l16;

  v8f acc[12];
#pragma unroll
  for (int i = 0; i < 12; ++i) acc[i] = (v8f){0.f,0.f,0.f,0.f,0.f,0.f,0.f,0.f};

#pragma unroll
  for (int kk = 0; kk < 6; ++kk) {
    int jA = kk * 4 + 2 * half;
    v2f bw;
    bw.x = weights[vc * J_ + jA];
    bw.y = weights[vc * J_ + jA + 1];
#pragma unroll
    for (int c = 0; c < 12; ++c) {
      v2f a;
      a.x = A_kcb[(jA * 16 + c) * B_ + bA];
      a.y = A_kcb[((jA + 1) * 16 + c) * B_ + bA];
      acc[c] = __builtin_amdgcn_wmma_f32_16x16x4_f32(false, a, false, bw,
                                                     (short)0, acc[c],
                                                     false, false);
    }
  }

  if (vOK) {
#pragma unroll
    for (int r = 0; r < 8; ++r) {
      int b = m0 + r + 8 * half;
      float px = v_posed[b * V3_ + v * 3 + 0];
      float py = v_posed[b * V3_ + v * 3 + 1];
      float pz = v_posed[b * V3_ + v * 3 + 2];
#pragma unroll
      for (int c = 0; c < 3; ++c) {
        float val = acc[c * 4 + 0][r] * px + acc[c * 4 + 1][r] * py +
                    acc[c * 4 + 2][r] * pz + acc[c * 4 + 3][r];
        verts[(b * V_ + v) * 3 + c] = val;
      }
    }
  }
}

// ---------------- 6) joint regression ----------------
__global__ void joints_kernel(const float* __restrict__ verts,
                              const float* __restrict__ JR19,
                              float* __restrict__ joints) {
  __shared__ float lds[256 * 57];
  const int b = blockIdx.x;
  const int t = threadIdx.x;
  float acc[57];
#pragma unroll
  for (int i = 0; i < 57; ++i) acc[i] = 0.f;
  for (int v = t; v < V_; v += 256) {
    float vx = verts[(b * V_ + v) * 3 + 0];
    float vy = verts[(b * V_ + v) * 3 + 1];
    float vz = verts[(b * V_ + v) * 3 + 2];
#pragma unroll
    for (int jj = 0; jj < 19; ++jj) {
      float w = JR19[v * 19 + jj];
      acc[jj * 3 + 0] += w * vx;
      acc[jj * 3 + 1] += w * vy;
      acc[jj * 3 + 2] += w * vz;
    }
  }
#pragma unroll
  for (int i = 0; i < 57; ++i) lds[t * 57 + i] = acc[i];
  __syncthreads();
  for (int s = 128; s > 0; s >>= 1) {
    if (t < s)
      for (int i = 0; i < 57; ++i) lds[t * 57 + i] += lds[(t + s) * 57 + i];
    __syncthreads();
  }
  if (t < 57) joints[b * 57 + t] = lds[t];
}

// ---------------- launcher ----------------
extern "C" void kernel_launch(void* const* d_in, const int* in_sizes, int n_in,
                              void* d_out, int out_size, void* d_ws, size_t ws_size,
                              hipStream_t stream) {
  const float* shape_params = (const float*)d_in[0];   // B x 10
  const float* pose_params  = (const float*)d_in[1];   // B x 72
  const float* v_template   = (const float*)d_in[2];   // V x 3
  const float* shapedirs    = (const float*)d_in[3];   // 10 x V3
  const float* J_regressor  = (const float*)d_in[4];   // V x 24
  const float* posedirs     = (const float*)d_in[5];   // 207 x V3
  const float* weights      = (const float*)d_in[6];   // V x 24
  const float* joint_reg    = (const float*)d_in[7];   // V x 19
  const int*   parents      = (const int*)d_in[8];     // 24

  float* ws = (float*)d_ws;
  float* Rs    = ws + RS_OFF;
  float* pf    = ws + PF_OFF;
  float* sp12  = ws + SP12_OFF;
  float* vp    = ws + VP_OFF;
  float* SD_J  = ws + SDJ_OFF;
  float* T_J   = ws + TJ_OFF;
  float* A_kcb = ws + AK_OFF;

  float* verts  = (float*)d_out;                       // B*V*3
  float* joints = verts + (size_t)B_ * V_ * 3;         // B*19*3
  float* J_out  = joints + (size_t)B_ * 19 * 3;        // B*24*3

  rodrigues_kernel<<<(B_ * J_ + 255) / 256, 256, 0, stream>>>(pose_params,
                                                              shape_params,
                                                              Rs, pf, sp12);
  sdj_kernel<<<72, 256, 0, stream>>>(shapedirs, v_template, J_regressor, SD_J, T_J);
  vposed_gemm_kernel<<<dim3(162, 64), 256, 0, stream>>>(sp12, pf, shapedirs,
                                                        posedirs, v_template, vp);
  rigid_kernel<<<B_, 32, 0, stream>>>(shape_params, Rs, SD_J, T_J, parents,
                                      A_kcb, J_out);
  skin_kernel<<<dim3(54, 64), 256, 0, stream>>>(A_kcb, weights, vp, verts);
  joints_kernel<<<B_, 256, 0, stream>>>(verts, joint_reg, joints);
}